// KANLayerEfficient_14714557956115
// MI455X (gfx1250) — compile-verified
//
#include <hip/hip_runtime.h>
#include <hip/hip_bf16.h>

// Problem constants (match reference: B=1024, IN=256, OUT=256, G=16)
#define KAN_B   1024
#define KAN_IN  256
#define KAN_OUT 256
#define KAN_G   16

#define BT 32            // batch rows per block (one per lane of a wave32)
#define OT 4             // output features per block (one per wave)
#define NT (BT * OT)     // 128 threads = 4 wave32
#define XPAD (KAN_IN + 1)  // padded LDS row stride for x (conflict-free column reads)

// ---- CDNA5 async global->LDS path (guarded; falls back to plain copies) ----
#if defined(__HIP_DEVICE_COMPILE__) &&                                   \
    __has_builtin(__builtin_amdgcn_global_load_async_to_lds_b32) &&      \
    __has_builtin(__builtin_amdgcn_global_load_async_to_lds_b128) &&     \
    __has_builtin(__builtin_amdgcn_s_wait_asynccnt)
#define USE_ASYNC_COPY 1
#else
#define USE_ASYNC_COPY 0
#endif

#if USE_ASYNC_COPY
#define GLOBAL_AS __attribute__((address_space(1)))
#define LDS_AS    __attribute__((address_space(3)))
typedef __attribute__((ext_vector_type(4))) int kan_v4i;

__device__ __forceinline__ void kan_async_b32(const float* g, float* l) {
    __builtin_amdgcn_global_load_async_to_lds_b32(
        (GLOBAL_AS int*)g, (LDS_AS int*)l, /*offset=*/0, /*cpol=*/0);
}
__device__ __forceinline__ void kan_async_b128(const float* g, float* l) {
    __builtin_amdgcn_global_load_async_to_lds_b128(
        (GLOBAL_AS kan_v4i*)g, (LDS_AS kan_v4i*)l, /*offset=*/0, /*cpol=*/0);
}
#endif

__device__ __forceinline__ float kan_clamp1(float v) {
#if defined(__HIP_DEVICE_COMPILE__)
    return __builtin_amdgcn_fmed3f(v, -1.0f, 1.0f);   // v_med3_f32: one-instruction clamp
#else
    return fminf(fmaxf(v, -1.0f), 1.0f);
#endif
}

__device__ __forceinline__ int kan_wave_id(int t) {
#if defined(__HIP_DEVICE_COMPILE__)
    return __builtin_amdgcn_readfirstlane(t >> 5);    // provably wave-uniform -> SGPR bases
#else
    return t >> 5;
#endif
}

__global__ __launch_bounds__(NT) void kan_layer_kernel(
    const float* __restrict__ x,      // [B, IN]
    const float* __restrict__ W,      // [OUT, IN]
    const float* __restrict__ S,      // [OUT, IN, G]
    const float* __restrict__ bias,   // [OUT]
    float* __restrict__ out)          // [B, OUT]
{
    __shared__ float s_sp[OT * KAN_IN * KAN_G]; // 64 KB  spline tile (contiguous chunk of S)
    __shared__ float s_x[BT * XPAD];            // ~32 KB padded x tile
    __shared__ float s_w[OT * KAN_IN];          // 4 KB   W rows for this block's outputs

    const int t  = (int)threadIdx.x;
    const int b0 = (int)blockIdx.x * BT;
    const int o0 = (int)blockIdx.y * OT;

    const float* xg = x + (size_t)b0 * KAN_IN;
    const float* sg = S + (size_t)o0 * (KAN_IN * KAN_G);
    const float* wg = W + (size_t)o0 * KAN_IN;

#if USE_ASYNC_COPY
    // x tile: 32 rows x 256 floats, LDS row stride 257 (pad) -> b32 granularity.
    // Consecutive lanes -> consecutive i: global coalesced, LDS conflict-free.
    #pragma unroll
    for (int k = 0; k < (BT * KAN_IN) / NT; ++k) {            // 64 iters
        int e = k * NT + t;                                   // 0..8191
        int b = e >> 8, i = e & (KAN_IN - 1);
        kan_async_b32(xg + b * KAN_IN + i, s_x + b * XPAD + i);
    }
    // spline tile: contiguous 64 KB, 16B aligned -> b128.
    #pragma unroll
    for (int k = 0; k < (OT * KAN_IN * KAN_G) / (4 * NT); ++k) { // 32 iters
        int e = k * NT + t;
        kan_async_b128(sg + 4 * e, s_sp + 4 * e);
    }
    // W tile: contiguous 4 KB -> b128.
    #pragma unroll
    for (int k = 0; k < (OT * KAN_IN) / (4 * NT); ++k) {      // 2 iters
        int e = k * NT + t;
        kan_async_b128(wg + 4 * e, s_w + 4 * e);
    }
    __builtin_amdgcn_s_wait_asynccnt(0);
    __syncthreads();
#else
    for (int k = 0; k < (BT * KAN_IN) / NT; ++k) {
        int e = k * NT + t;
        int b = e >> 8, i = e & (KAN_IN - 1);
        s_x[b * XPAD + i] = xg[b * KAN_IN + i];
    }
    for (int k = 0; k < (OT * KAN_IN * KAN_G) / (4 * NT); ++k) {
        int e = k * NT + t;
        reinterpret_cast<float4*>(s_sp)[e] = reinterpret_cast<const float4*>(sg)[e];
    }
    for (int k = 0; k < (OT * KAN_IN) / (4 * NT); ++k) {
        int e = k * NT + t;
        reinterpret_cast<float4*>(s_w)[e] = reinterpret_cast<const float4*>(wg)[e];
    }
    __syncthreads();
#endif

    const int wave = kan_wave_id(t);      // 0..3  -> output feature (wave-uniform)
    const int lane = t & 31;              // 0..31 -> batch row (wave32)

    const float* xr = s_x + lane * XPAD;               // conflict-free column reads
    const float* wr = s_w + wave * KAN_IN;             // uniform addr -> LDS broadcast
    const float* sr = s_sp + wave * (KAN_IN * KAN_G);  // gather base for this wave's o

    float acc = 0.0f;
    #pragma unroll 8
    for (int i = 0; i < KAN_IN; ++i) {
        float xv = xr[i];
        float wv = wr[i];
        float tc = kan_clamp1(xv * wv);                        // v_med3_f32
        float u  = __builtin_fmaf(tc, 7.5f, 7.5f);             // (t+1)*(G-1)/2, u in [0,15]
        float fl = __builtin_fminf(__builtin_floorf(u), 14.0f);// left knot (clamped to G-2)
        int   idx = (int)fl;                                   // u>=0 -> trunc == floor
        float fr  = u - fl;                                    // frac (==1 when u hits 15)
        float v0  = sr[(i << 4) + idx];                        // LDS gather
        float v1  = sr[(i << 4) + idx + 1];
        acc = __builtin_fmaf(fr, v1 - v0, acc + v0);           // acc += v0 + fr*(v1-v0)
    }

    out[(size_t)(b0 + lane) * KAN_OUT + (o0 + wave)] = acc + bias[o0 + wave];
}

extern "C" void kernel_launch(void* const* d_in, const int* in_sizes, int n_in,
                              void* d_out, int out_size, void* d_ws, size_t ws_size,
                              hipStream_t stream) {
    (void)in_sizes; (void)n_in; (void)out_size; (void)d_ws; (void)ws_size;
    const float* x    = (const float*)d_in[0];
    const float* W    = (const float*)d_in[1];
    const float* S    = (const float*)d_in[2];
    const float* bias = (const float*)d_in[3];
    float* out = (float*)d_out;

    dim3 grid(KAN_B / BT, KAN_OUT / OT);   // (32, 64)
    kan_layer_kernel<<<grid, NT, 0, stream>>>(x, W, S, bias, out);
}